// bicon_loss_71837622993225
// MI455X (gfx1250) — compile-verified
//
#include <hip/hip_runtime.h>
#include <hip/hip_bf16.h>

// CDNA5 / gfx1250, wave32.
typedef __attribute__((ext_vector_type(2))) float v2f;
typedef __attribute__((ext_vector_type(8))) float v8f;

#define BATCH 16
#define HH    512
#define WW    512
#define TW    64            // tile width  (pixels), 4 per thread
#define TH    16            // tile height (pixels)
#define SROW  (TW + 4)      // LDS row stride in floats (68 -> 272B, 16B aligned)
#define HALOW (TW + 2)      // 66 halo columns
#define HALOH (TH + 2)      // 18 halo rows

__device__ __forceinline__ float softplus_f(float x) {
    // stable log(1+e^x); __expf -> v_exp_f32
    return fmaxf(x, 0.0f) + log1pf(__expf(-fabsf(x)));
}

// Wave32 sum-reduction on the matrix pipe: one v_wmma_f32_16x16x4_f32.
// A[m,k]: lanes 0-15 VGPR0 -> K=0, lanes 16-31 VGPR0 -> K=2 (VGPR1 zeroed).
// With B = ones, D[m,n] = v_m + v_{m+16}. Summing D's 8 VGPRs per lane gives
// S_low (lanes 0-15) / S_high (lanes 16-31); total = S_low + S_high.
// EXEC must be all ones at the call site.
__device__ __forceinline__ float wave_reduce_wmma(float v) {
    v2f a; a[0] = v;    a[1] = 0.0f;
    v2f b; b[0] = 1.0f; b[1] = 1.0f;
    v8f c = {};
    c = __builtin_amdgcn_wmma_f32_16x16x4_f32(false, a, false, b,
                                              (short)0, c, false, false);
    float s = 0.0f;
#pragma unroll
    for (int i = 0; i < 8; ++i) s += c[i];
    float hi = __shfl(s, 16, 32);
    return s + hi;   // full-wave sum valid in lanes 0..15 (lane 0 used)
}

__global__ __launch_bounds__(256) void bicon_partial_kernel(
    const float* __restrict__ cmap,     // [16,2,512,512]
    const int*   __restrict__ target,   // [16,512,512]
    const float* __restrict__ con,      // [16,8,512,512]
    float*       __restrict__ partial)
{
    __shared__ float s0[HALOH * SROW];
    __shared__ float s1[HALOH * SROW];
    __shared__ float wred[8];

    const int tid = threadIdx.x;
    const int b  = blockIdx.z;
    const int h0 = blockIdx.y * TH;
    const int w0 = blockIdx.x * TW;
    const size_t plane = (size_t)HH * WW;
    const float* c0p = cmap + ((size_t)b * 2 + 0) * plane;
    const float* c1p = cmap + ((size_t)b * 2 + 1) * plane;

    // Stage 18x66 halo tile for both classes into LDS (zero-padded borders).
    for (int idx = tid; idx < HALOH * HALOW; idx += 256) {
        const int r = idx / HALOW;
        const int cc = idx % HALOW;
        const int lh = h0 - 1 + r;
        const int lw = w0 - 1 + cc;
        const bool in = (lh >= 0) & (lh < HH) & (lw >= 0) & (lw < WW);
        const size_t off = (size_t)lh * WW + lw;
        s0[r * SROW + cc] = in ? c0p[off] : 0.0f;
        s1[r * SROW + cc] = in ? c1p[off] : 0.0f;
    }
    __syncthreads();

    // 4 consecutive pixels per thread along W: wbase..wbase+3.
    const int tx = tid & 15;            // 16 thread-columns * 4 px = 64
    const int ty = tid >> 4;            // 16 rows
    const int h = h0 + ty;
    const int wbase = w0 + 4 * tx;
    const int lbase = (ty + 1) * SROW + 1 + 4 * tx;

    float cen0[4], cen1[4];
#pragma unroll
    for (int j = 0; j < 4; ++j) {
        cen0[j] = s0[lbase + j];
        cen1[j] = s1[lbase + j];
    }

    // Cross-entropy: lse - c_t == softplus(c_other - c_t). int4 -> b128 load.
    const int4 t4 = *(const int4*)(target + (size_t)b * plane
                                          + (size_t)h * WW + wbase);
    const int tt[4] = {t4.x, t4.y, t4.z, t4.w};
    float cesum = 0.0f;
#pragma unroll
    for (int j = 0; j < 4; ++j)
        cesum += softplus_f((tt[j] == 0) ? (cen1[j] - cen0[j])
                                         : (cen0[j] - cen1[j]));

    // 8 directions; shifted[i](h,w) = c_map(h + dhs[i], w + dws[i]).
    const int dhs[8] = {-1, -1, -1,  0, 0,  1, 1, 1};
    const int dws[8] = {-1,  0,  1, -1, 1, -1, 0, 1};

    const float* conb = con + (size_t)b * 8 * plane + (size_t)h * WW + wbase;
    float accsum = 0.0f;
#pragma unroll
    for (int i = 0; i < 8; ++i) {
        const float4 ct4 = *(const float4*)(conb + (size_t)i * plane); // b128
        const float cta[4] = {ct4.x, ct4.y, ct4.z, ct4.w};
        const int nb = (ty + 1 + dhs[i]) * SROW + 1 + 4 * tx + dws[i];
#pragma unroll
        for (int j = 0; j < 4; ++j) {
            const float n0 = s0[nb + j];
            const float n1 = s1[nb + j];
            const float d = cen0[j] * n0 - cen1[j] * n1;
            // edge = ((2ct < 8) & (2ct > 0))
            const float edge = (cta[j] > 0.0f && cta[j] < 4.0f) ? 1.0f : 0.0f;
            // min(softmax2) = sigmoid(-|d|); -log(1-p) = softplus(-|d|)
            float l = log1pf(__expf(-fabsf(d)));
            l = fminf(l, 100.0f);   // torch log clamp (never triggers)
            accsum += edge * l;
        }
    }

    // Fold loss weights and mean divisors per pixel.
    const float W_CE  = 0.8f / (float)((size_t)BATCH * HH * WW);      // /2^22
    const float W_MIN = 0.2f / (float)((size_t)BATCH * 8 * HH * WW);  // /2^25
    const float v = W_CE * cesum + W_MIN * accsum;

    // Wave32 reduce on the matrix pipe, then cross-wave via LDS.
    const float wsum = wave_reduce_wmma(v);
    if ((tid & 31) == 0) wred[tid >> 5] = wsum;
    __syncthreads();
    if (tid == 0) {
        float tblk = 0.0f;
#pragma unroll
        for (int i = 0; i < 8; ++i) tblk += wred[i];
        const size_t bid = ((size_t)blockIdx.z * gridDim.y + blockIdx.y) * gridDim.x
                           + blockIdx.x;
        partial[bid] = tblk;
    }
}

__global__ __launch_bounds__(256) void bicon_final_kernel(
    const float* __restrict__ partial, int n, float* __restrict__ out)
{
    __shared__ float wred[8];
    const int tid = threadIdx.x;
    float s = 0.0f;
    // n = 4096 = 256 * 16: read as float4 for b128 loads too.
    const float4* p4 = (const float4*)partial;
    for (int i = tid; i < (n >> 2); i += 256) {
        const float4 v = p4[i];
        s += (v.x + v.y) + (v.z + v.w);
    }
    const float wsum = wave_reduce_wmma(s);
    if ((tid & 31) == 0) wred[tid >> 5] = wsum;
    __syncthreads();
    if (tid == 0) {
        float tot = 0.0f;
#pragma unroll
        for (int i = 0; i < 8; ++i) tot += wred[i];
        out[0] = tot;
    }
}

extern "C" void kernel_launch(void* const* d_in, const int* in_sizes, int n_in,
                              void* d_out, int out_size, void* d_ws, size_t ws_size,
                              hipStream_t stream) {
    const float* cmap   = (const float*)d_in[0];  // [16,2,512,512] f32
    const int*   target = (const int*)d_in[1];    // [16,512,512] int
    const float* con    = (const float*)d_in[2];  // [16,8,512,512] f32
    float* out     = (float*)d_out;               // scalar f32
    float* partial = (float*)d_ws;                // 4096 floats = 16 KB

    dim3 grid(WW / TW, HH / TH, BATCH);           // (8, 32, 16) = 4096 blocks
    bicon_partial_kernel<<<grid, 256, 0, stream>>>(cmap, target, con, partial);

    const int nblocks = (WW / TW) * (HH / TH) * BATCH;
    bicon_final_kernel<<<1, 256, 0, stream>>>(partial, nblocks, out);
}